// ChebConvolution_83288005804107
// MI455X (gfx1250) — compile-verified
//
#include <hip/hip_runtime.h>

typedef float v2f __attribute__((ext_vector_type(2)));
typedef float v8f __attribute__((ext_vector_type(8)));

#define N_NODES 50000
#define DIM     128
#define KPAIRS  (DIM / 2)                     // 64 K-pairs
#define ROWS_PER_BLOCK 128                    // 8 waves x 16-row M-tile each
#define GEMM_GRID ((N_NODES + ROWS_PER_BLOCK - 1) / ROWS_PER_BLOCK)   // 391

// ---------------------------------------------------------------------------
// GEMM: Out[M,128] = A[M,128] @ W[128,128] via V_WMMA_F32_16X16X4_F32.
// Block = 256 threads = 8 waves. Wave w owns M-tile rows [blk*128+16w, +16)
// and sweeps all 8 column tiles with 8 accumulators: one A fragment feeds
// 8 WMMAs; A is read from L2 exactly once per GEMM.
// W is staged in LDS PAIR-MAJOR:  Ws2[j*128+n] = {W[2j][n], W[2j+1][n]}
// so every B fragment is one aligned ds_load_b64 (no re-pairing moves).
// ---------------------------------------------------------------------------
__global__ __launch_bounds__(256) void gemm_f32_wmma(const float* __restrict__ A,
                                                     const float* __restrict__ W,
                                                     float* __restrict__ Out,
                                                     int nrows) {
    __shared__ v2f Ws2[KPAIRS * DIM];         // 64 KB of the 320 KB WGP LDS
    for (int idx = threadIdx.x; idx < KPAIRS * DIM; idx += 256) {
        const int j = idx >> 7;               // K-pair index
        const int n = idx & (DIM - 1);        // column
        v2f p;
        p.x = W[(size_t)(2 * j)     * DIM + n];
        p.y = W[(size_t)(2 * j + 1) * DIM + n];
        Ws2[idx] = p;                         // ds_store_b64
    }
    __syncthreads();

    const int wave = threadIdx.x >> 5;
    const int lane = threadIdx.x & 31;
    const int l16  = lane & 15;               // A row-in-tile / B,C column-in-tile
    const int kh   = lane >> 4;               // K half-select (WMMA f32 A/B layout)
    const int mb   = blockIdx.x * ROWS_PER_BLOCK + wave * 16;
    if (mb >= nrows) return;                  // wave-uniform: EXEC all-1 inside

    const float* arow  = A   + (size_t)(mb + l16) * DIM + 2 * kh;
    const v2f*   bbase = Ws2 + (size_t)kh * DIM + l16;   // K-pair j = 2*ks + kh

    v8f c[8];
    #pragma unroll
    for (int t = 0; t < 8; ++t) c[t] = (v8f){0, 0, 0, 0, 0, 0, 0, 0};

    #pragma unroll 8
    for (int ks = 0; ks < DIM / 4; ++ks) {
        v2f a = *(const v2f*)(arow + 4 * ks);           // global_load_b64
        const v2f* brow = bbase + (size_t)(2 * ks) * DIM;
        #pragma unroll
        for (int t = 0; t < 8; ++t) {                   // sweep 8 column tiles
            v2f b = brow[t * 16];                       // one ds_load_b64
            c[t] = __builtin_amdgcn_wmma_f32_16x16x4_f32(
                       false, a, false, b, (short)0, c[t], false, false);
        }
    }

    // C/D layout: VGPR j -> row mb + j + 8*kh, column t*16 + l16
    float* obase = Out + (size_t)(mb + 8 * kh) * DIM + l16;
    #pragma unroll
    for (int t = 0; t < 8; ++t) {
        #pragma unroll
        for (int j = 0; j < 8; ++j) obase[(size_t)j * DIM + t * 16] = c[t][j];
    }
}

// ---------------------------------------------------------------------------
// S = -Src   (accumulator init with the "- x_{i-2}" Chebyshev term)
// ---------------------------------------------------------------------------
__global__ __launch_bounds__(256) void neg_copy(const float4* __restrict__ src,
                                                float4* __restrict__ dst, int n4) {
    int i = blockIdx.x * 256 + threadIdx.x;
    if (i < n4) {
        float4 v = src[i];
        dst[i] = make_float4(-v.x, -v.y, -v.z, -v.w);
    }
}

// ---------------------------------------------------------------------------
// S[row] += 2*val * X[col] per edge; 128 lanes = feature dims. fp32 atomics
// resolve at L2 (low contention: 102M adds over 6.4M addresses; the whole
// working set is resident in the 192 MB L2).
// ---------------------------------------------------------------------------
#define EDGES_PER_BLOCK 8
__global__ __launch_bounds__(128) void spmm_atomic(const int* __restrict__ rows,
                                                   const int* __restrict__ cols,
                                                   const float* __restrict__ vals,
                                                   const float* __restrict__ X,
                                                   float* __restrict__ S, int E) {
    const int d  = threadIdx.x;
    const int e0 = blockIdx.x * EDGES_PER_BLOCK;
    const int e1 = min(e0 + EDGES_PER_BLOCK, E);
    for (int e = e0; e < e1; ++e) {
        const int   r = rows[e];
        const int   c = cols[e];
        const float v = 2.0f * vals[e];
        __hip_atomic_fetch_add(&S[(size_t)r * DIM + d], v * X[(size_t)c * DIM + d],
                               __ATOMIC_RELAXED, __HIP_MEMORY_SCOPE_AGENT);
    }
}

// ---------------------------------------------------------------------------
extern "C" void kernel_launch(void* const* d_in, const int* in_sizes, int n_in,
                              void* d_out, int out_size, void* d_ws, size_t ws_size,
                              hipStream_t stream) {
    const float* X    = (const float*)d_in[0];
    const int*   rows = (const int*)d_in[1];
    const int*   cols = (const int*)d_in[2];
    const float* vals = (const float*)d_in[3];
    const float* W    = (const float*)d_in[4];
    const int    E    = in_sizes[1];

    const size_t mat = (size_t)N_NODES * DIM;          // 6.4M floats / 25.6 MB
    float* XW  = (float*)d_ws;                         // x0 == x1
    float* S   = XW + mat;                             // spmm accumulator
    float* X2  = S + mat;                              // x2
    float* out = (float*)d_out;

    const int n4 = (int)(mat / 4);
    const dim3 gB(256), gG(GEMM_GRID);
    const dim3 nB(256), nG((n4 + 255) / 256);
    const dim3 sB(128), sG((E + EDGES_PER_BLOCK - 1) / EDGES_PER_BLOCK);

    // x0 = x1 = XW = X @ W
    gemm_f32_wmma<<<gG, gB, 0, stream>>>(X, W, XW, N_NODES);

    // S = spmm(2A, x1) - x0 ; x2 = S @ W
    neg_copy<<<nG, nB, 0, stream>>>((const float4*)XW, (float4*)S, n4);
    spmm_atomic<<<sG, sB, 0, stream>>>(rows, cols, vals, XW, S, E);
    gemm_f32_wmma<<<gG, gB, 0, stream>>>(S, W, X2, N_NODES);

    // S = spmm(2A, x2) - x1 ; out = S @ W
    neg_copy<<<nG, nB, 0, stream>>>((const float4*)XW, (float4*)S, n4);
    spmm_atomic<<<sG, sB, 0, stream>>>(rows, cols, vals, X2, S, E);
    gemm_f32_wmma<<<gG, gB, 0, stream>>>(S, W, out, N_NODES);
}